// Net_R1L_40295383171500
// MI455X (gfx1250) — compile-verified
//
#include <hip/hip_runtime.h>
#include <hip/hip_bf16.h>
#include <math.h>

typedef __attribute__((ext_vector_type(2))) float v2f;
typedef __attribute__((ext_vector_type(8))) float v8f;

// Single-VALU ReLU: v_max_num_f32 dst, 0, src. (fmaxf/fmed3 both lower to a
// 2-op canonicalize+max sequence; WMMA outputs are never sNaN so the
// canonicalize is unnecessary, and this op sits on the serial recurrence path.)
__device__ __forceinline__ float relu1(float x) {
    float r;
    asm("v_max_num_f32 %0, 0, %1" : "=v"(r) : "v"(x));
    return r;
}

// ReLU-RNN recurrence on V_WMMA_F32_16X16X4_F32 (full f32 precision).
// One wave handles TWO groups of 16 batch chains (32 chains/wave, 128 waves):
// the two groups' WMMA chains are independent, letting the scheduler fill the
// WMMA->WMMA latency bubbles of one group with the other group's work.
// D/C matrix per group: rows = hidden unit (0..4), cols = chain. Rows 5..15
// are invariantly zero, so post-ReLU h-registers are directly valid B operands
// (K rows in low lane half, zeros in high half) -- no cross-lane traffic.
__global__ __launch_bounds__(256) void rnn_wmma_kernel(
    const float* __restrict__ state,
    const float* __restrict__ W_ih, const float* __restrict__ W_hh,
    const float* __restrict__ b_ih, const float* __restrict__ b_hh,
    const float* __restrict__ W_out, const float* __restrict__ b_out,
    float* __restrict__ out, int B, int T)
{
    const int lane = threadIdx.x & 31;      // wave32
    const int m    = lane & 15;             // A-matrix row / chain-in-group
    const bool low = lane < 16;
    const int w    = blockIdx.x * (blockDim.x >> 5) + (threadIdx.x >> 5);
    if (w * 32 >= B) return;                // uniform per wave: EXEC stays all-1s

    // Guarded per-lane gathers of the tiny weight matrices (done once).
    auto whh = [&](int r, int c) -> float { return (r < 5) ? W_hh[r * 5 + c] : 0.0f; };
    auto wih = [&](int r, int c) -> float { return (r < 5) ? W_ih[r * 3 + c] : 0.0f; };

    // A operands (16x4 f32): VGPR0 = {K0 low half, K2 high half}, VGPR1 = {K1, K3}.
    // WMMA-a: K0,K1 = W_hh cols 0,1 ; WMMA-b: K0,K1 = W_hh cols 2,3
    // WMMA-c: K0 = W_hh col 4, K1 = W_ih col 0, K2 = W_ih col 1, K3 = W_ih col 2
    // Shared by both chain-groups.
    v2f Aa, Ab, Ac;
    Aa.x = low ? whh(m, 0) : 0.0f;
    Aa.y = low ? whh(m, 1) : 0.0f;
    Ab.x = low ? whh(m, 2) : 0.0f;
    Ab.y = low ? whh(m, 3) : 0.0f;
    Ac.x = low ? whh(m, 4) : wih(m, 1);
    Ac.y = low ? wih(m, 0) : wih(m, 2);

    // Constant bias accumulator: row j = b_ih[j]+b_hh[j] (j<5), rows 5..15 = 0.
    v8f cbias;
#pragma unroll
    for (int j = 0; j < 8; ++j) cbias[j] = 0.0f;
#pragma unroll
    for (int j = 0; j < 5; ++j) cbias[j] = low ? (b_ih[j] + b_hh[j]) : 0.0f;

    // Per-group state.
    float h[2][5];
    const float* p[2];
    float4 c0[2], c1[2], c2[2];
#pragma unroll
    for (int g = 0; g < 2; ++g) {
#pragma unroll
        for (int j = 0; j < 5; ++j) h[g][j] = 0.0f;
        const int chain = w * 32 + g * 16 + m;
        p[g] = state + (size_t)chain * (size_t)T * 3u;   // 24 KB/chain, 16B aligned
        c0[g] = ((const float4*)p[g])[0];
        c1[g] = ((const float4*)p[g])[1];
        c2[g] = ((const float4*)p[g])[2];
    }

    auto step = [&](int g, float x0, float x1, float x2) {
        v2f Ba, Bb, Bc;
        Ba.x = h[g][0];            // B rows K0/K2 (high half zero)
        Ba.y = h[g][1];            // K1/K3
        Bb.x = h[g][2];
        Bb.y = h[g][3];
        Bc.x = low ? h[g][4] : x1; // K0 = h4, K2 = x1 (upper lanes hold same chain's x)
        Bc.y = low ? x0 : x2;      // K1 = x0, K3 = x2
        v8f acc = __builtin_amdgcn_wmma_f32_16x16x4_f32(false, Aa, false, Ba,
                                                        (short)0, cbias, false, false);
        acc = __builtin_amdgcn_wmma_f32_16x16x4_f32(false, Ab, false, Bb,
                                                    (short)0, acc, false, false);
        acc = __builtin_amdgcn_wmma_f32_16x16x4_f32(false, Ac, false, Bc,
                                                    (short)0, acc, false, false);
        h[g][0] = relu1(acc[0]);
        h[g][1] = relu1(acc[1]);
        h[g][2] = relu1(acc[2]);
        h[g][3] = relu1(acc[3]);
        h[g][4] = relu1(acc[4]);
    };

    // 4 timesteps = 48 B/group = 3x b128 loads (16B aligned). Main loop runs
    // T/4-1 iterations with unconditional next-block loads (always in-bounds);
    // the final block is processed after the loop. Speculative prefetch
    // (global_prefetch_b8, SE scope) may overshoot by <=384B; the ISA drops
    // failed speculative translations silently.
    const int nblk = T >> 2;
    for (int it = 0; it < nblk - 1; ++it) {
        float4 n0[2], n1[2], n2[2];
#pragma unroll
        for (int g = 0; g < 2; ++g) {
            const float4* pn = (const float4*)(p[g] + 12);
            n0[g] = pn[0]; n1[g] = pn[1]; n2[g] = pn[2];
            __builtin_prefetch(p[g] + 96, 0, 3);   // ~32 timesteps ahead
        }

        // Interleave the two independent chains at each timestep.
        step(0, c0[0].x, c0[0].y, c0[0].z);  step(1, c0[1].x, c0[1].y, c0[1].z);
        step(0, c0[0].w, c1[0].x, c1[0].y);  step(1, c0[1].w, c1[1].x, c1[1].y);
        step(0, c1[0].z, c1[0].w, c2[0].x);  step(1, c1[1].z, c1[1].w, c2[1].x);
        step(0, c2[0].y, c2[0].z, c2[0].w);  step(1, c2[1].y, c2[1].z, c2[1].w);

#pragma unroll
        for (int g = 0; g < 2; ++g) {
            c0[g] = n0[g]; c1[g] = n1[g]; c2[g] = n2[g];
            p[g] += 12;
        }
    }
    // Final 4 timesteps (no further loads).
    step(0, c0[0].x, c0[0].y, c0[0].z);  step(1, c0[1].x, c0[1].y, c0[1].z);
    step(0, c0[0].w, c1[0].x, c1[0].y);  step(1, c0[1].w, c1[1].x, c1[1].y);
    step(0, c1[0].z, c1[0].w, c2[0].x);  step(1, c1[1].z, c1[1].w, c2[1].x);
    step(0, c2[0].y, c2[0].z, c2[0].w);  step(1, c2[1].y, c2[1].z, c2[1].w);

    // Epilogue: lane n<16 holds h[0..4] of its chains. out = tanh(W_out.h + b_out)
    if (low) {
#pragma unroll
        for (int g = 0; g < 2; ++g) {
            float z = W_out[0] * h[g][0] + W_out[1] * h[g][1] + W_out[2] * h[g][2] +
                      W_out[3] * h[g][3] + W_out[4] * h[g][4] + b_out[0];
            out[w * 32 + g * 16 + m] = tanhf(z);
        }
    }
}

extern "C" void kernel_launch(void* const* d_in, const int* in_sizes, int n_in,
                              void* d_out, int out_size, void* d_ws, size_t ws_size,
                              hipStream_t stream) {
    (void)n_in; (void)out_size; (void)d_ws; (void)ws_size;
    const float* state = (const float*)d_in[0];
    const float* W_ih  = (const float*)d_in[1];
    const float* W_hh  = (const float*)d_in[2];
    const float* b_ih  = (const float*)d_in[3];
    const float* b_hh  = (const float*)d_in[4];
    const float* W_out = (const float*)d_in[5];
    const float* b_out = (const float*)d_in[6];
    float* out = (float*)d_out;

    const int T = 2048;
    const int B = in_sizes[0] / (T * 3);      // 4096
    const int waves = B / 32;                 // 32 chains per wave -> 128 waves
    const int wavesPerBlock = 8;              // 256 threads
    const int blocks = (waves + wavesPerBlock - 1) / wavesPerBlock;

    rnn_wmma_kernel<<<blocks, wavesPerBlock * 32, 0, stream>>>(
        state, W_ih, W_hh, b_ih, b_hh, W_out, b_out, out, B, T);
}